// PointCNP_17952963297846
// MI455X (gfx1250) — compile-verified
//
#include <hip/hip_runtime.h>
#include <hip/hip_bf16.h>

// ---------------------------------------------------------------------------
// PointCNP forward for MI455X (gfx1250, wave32).
// Pipeline:
//   k_zero       : zero the (64,784,784) Sigma output (157MB, float4 stores)
//   k_neighbors  : fixed-grid top-25 kNN -> idx[784][25], offsets[784][25][2]
//   k_wmlp       : per-layer weight MLP (2->32->32->16), batch-independent
//   k_pointconv  : per grid point n: out[b,o] = sum_{k,c} f[b,idx[n,k],c]*G[(k,c),o]
//                  via V_WMMA_F32_16X16X4_F32 (M=cout, K=25*cin chunked, N=batch)
//                  operand staging via GLOBAL_LOAD_ASYNC_TO_LDS_B32 (ASYNCcnt)
//   k_musig      : rbf readout -> mu (64,784) and diag of Sigma
// ---------------------------------------------------------------------------

typedef float v2f __attribute__((ext_vector_type(2)));
typedef float v8f __attribute__((ext_vector_type(8)));

#define NPTS   784
#define NBHD_K 25
#define BATCH  64
#define NCTX   400

__device__ __forceinline__ float gcoord(int i) {
    return -14.0f + (float)i * (28.0f / 27.0f);   // linspace(-14,14,28)
}
__device__ __forceinline__ float swishf(float x) {
    return x / (1.0f + __expf(-x));
}
__device__ __forceinline__ float softplusf(float x) {
    return fmaxf(x, 0.0f) + log1pf(__expf(-fabsf(x)));
}

// ----------------------------- zero Sigma ----------------------------------
__global__ void k_zero(float4* __restrict__ p, long n4) {
    long i = (long)blockIdx.x * blockDim.x + threadIdx.x;
    long stride = (long)gridDim.x * blockDim.x;
    float4 z; z.x = 0.f; z.y = 0.f; z.z = 0.f; z.w = 0.f;
    for (; i < n4; i += stride) p[i] = z;
}

// -------------------------- fixed-grid kNN ---------------------------------
__global__ void k_neighbors(int* __restrict__ nidx, float* __restrict__ offs) {
    int n = blockIdx.x * blockDim.x + threadIdx.x;
    if (n >= NPTS) return;
    float xn = gcoord(n / 28), yn = gcoord(n % 28);
    float bd2[NBHD_K];
    int   bix[NBHD_K];
    for (int s = 0; s < NBHD_K; ++s) { bd2[s] = 1e30f; bix[s] = 0; }
    for (int m = 0; m < NPTS; ++m) {
        float dx = gcoord(m / 28) - xn;
        float dy = gcoord(m % 28) - yn;
        float d2 = dx * dx + dy * dy;
        if (d2 < bd2[NBHD_K - 1]) {          // stable insert (ties keep lower idx)
            int p = NBHD_K - 1;
            while (p > 0 && bd2[p - 1] > d2) {
                bd2[p] = bd2[p - 1]; bix[p] = bix[p - 1]; --p;
            }
            bd2[p] = d2; bix[p] = m;
        }
    }
    for (int s = 0; s < NBHD_K; ++s) {
        int m = bix[s];
        nidx[n * NBHD_K + s] = m;
        offs[(n * NBHD_K + s) * 2 + 0] = gcoord(m / 28) - xn;
        offs[(n * NBHD_K + s) * 2 + 1] = gcoord(m % 28) - yn;
    }
}

// ------------------------- weight-MLP (all layers) -------------------------
struct MLP4 {
    const float* w1[4]; const float* b1[4];
    const float* w2[4]; const float* b2[4];
    const float* w3[4]; const float* b3[4];
};

__global__ void k_wmlp(const float* __restrict__ offs, float* __restrict__ wmat, MLP4 P) {
    int t = blockIdx.x * blockDim.x + threadIdx.x;
    if (t >= 4 * NPTS * NBHD_K) return;
    int l = t / (NPTS * NBHD_K);
    int rem = t % (NPTS * NBHD_K);
    int n = rem / NBHD_K, k = rem % NBHD_K;
    float ox = offs[(n * NBHD_K + k) * 2 + 0];
    float oy = offs[(n * NBHD_K + k) * 2 + 1];
    const float *w1 = P.w1[l], *b1 = P.b1[l];
    const float *w2 = P.w2[l], *b2 = P.b2[l];
    const float *w3 = P.w3[l], *b3 = P.b3[l];
    float h1[32];
    #pragma unroll
    for (int e = 0; e < 32; ++e)
        h1[e] = swishf(w1[e * 2] * ox + w1[e * 2 + 1] * oy + b1[e]);
    float h2[32];
    #pragma unroll
    for (int e2 = 0; e2 < 32; ++e2) {
        float s = b2[e2];
        #pragma unroll
        for (int e = 0; e < 32; ++e) s += w2[e2 * 32 + e] * h1[e];
        h2[e2] = swishf(s);
    }
    float* wout = &wmat[((l * NPTS + n) * NBHD_K + k) * 16];
    #pragma unroll
    for (int e3 = 0; e3 < 16; ++e3) {
        float s = b3[e3];
        #pragma unroll
        for (int e = 0; e < 32; ++e) s += w3[e3 * 32 + e] * h2[e];
        wout[e3] = swishf(s);
    }
}

// ----------------------- SetConv encoding -> f0 ----------------------------
__global__ void k_h(const float* __restrict__ ctx_c, const float* __restrict__ ctx_v,
                    const float* __restrict__ kp, float* __restrict__ f0) {
    int t = blockIdx.x * blockDim.x + threadIdx.x;
    if (t >= BATCH * NPTS) return;
    int b = t / NPTS, n = t % NPTS;
    float x = gcoord(n / 28), y = gcoord(n % 28);
    float ls = kp[0], os = kp[1];
    float sc = -0.5f / (ls * ls);
    const float* cc = ctx_c + (size_t)b * NCTX * 2;
    const float* cv = ctx_v + (size_t)b * NCTX;
    float h0 = 0.f, h1 = 0.f;
    for (int j = 0; j < NCTX; ++j) {
        float dx = x - cc[2 * j], dy = y - cc[2 * j + 1];
        float psi = os * __expf(sc * (dx * dx + dy * dy));
        h0 += psi; h1 += psi * cv[j];
    }
    f0[t * 4 + 0] = x;
    f0[t * 4 + 1] = y;
    f0[t * 4 + 2] = h0;
    f0[t * 4 + 3] = h1 / (h0 + 1e-8f);
}

// ---------------------- PointConv layer (WMMA f32) -------------------------
// Per block (one per grid point n): out[b,o] = ((sum G*nv) + lb[o]) / 25.
// G[(k,c),o] = sum_e w[n,k,e]*lw[o, c*16+e], computed per K-chunk in LDS.
// WMMA mapping: M = cout (padded to 16), K = 25*cin (5 chunks of 5 nbrs), N = batch(64).
// Neighbor features staged global->LDS with GLOBAL_LOAD_ASYNC_TO_LDS_B32.
template <int CIN, int COUT, int ACT>
__global__ __launch_bounds__(256) void k_pointconv(
    const float* __restrict__ fin, float* __restrict__ fout,
    const int* __restrict__ nidx, const float* __restrict__ wmat,
    const float* __restrict__ lw, const float* __restrict__ lb)
{
    constexpr int KC = 5;
    constexpr int NCHUNK = NBHD_K / KC;
    constexpr int KSLAB = KC * CIN;
    constexpr int COUTP = (COUT < 16) ? 16 : COUT;
    constexpr int MT = COUTP / 16;
    constexpr int NT = BATCH / 16;          // 4
    constexpr int TILES = MT * NT;

    __shared__ float s_w[NBHD_K * 16];
    __shared__ float s_G[COUTP * KSLAB];
    __shared__ float s_nv[KSLAB * 65];      // stride 65 avoids bank conflicts

    const int n = blockIdx.x;
    const int tid = threadIdx.x;
    const int lane = tid & 31;
    const int wave = tid >> 5;
    const bool wact = wave < TILES;         // wave-uniform -> EXEC all-ones inside
    const int mt = wave / NT;
    const int nt = wave % NT;

    for (int i = tid; i < NBHD_K * 16; i += 256)
        s_w[i] = wmat[n * (NBHD_K * 16) + i];

    v8f acc = {0.f, 0.f, 0.f, 0.f, 0.f, 0.f, 0.f, 0.f};
    const int rowo  = mt * 16 + (lane & 15);       // A fragment: M index
    const int colb  = nt * 16 + (lane & 15);       // B fragment: N index (batch)
    const int khalf = (lane >> 4) * 2;             // K sub-offset per lane-half

    __syncthreads();

    for (int ch = 0; ch < NCHUNK; ++ch) {
        const int k0 = ch * KC;
        // --- gather neighbor features, transposed to [K][batch], via async
        //     global->LDS DMA (one GLOBAL_LOAD_ASYNC_TO_LDS_B32 per element;
        //     BATCH*KSLAB is an exact multiple of 256 so EXEC is all-ones) ---
        for (int p = tid; p < BATCH * KSLAB; p += 256) {
            const int b = p / KSLAB;
            const int kc = p % KSLAB;
            const int kl = kc / CIN;
            const int c = kc % CIN;
            const int nb = nidx[n * NBHD_K + k0 + kl];
            const float* gp = &fin[((size_t)b * NPTS + nb) * CIN + c];
            // low 32 bits of the flat shared address = LDS byte offset
            unsigned lds_addr = (unsigned)(size_t)(&s_nv[kc * 65 + b]);
            asm volatile("global_load_async_to_lds_b32 %0, %1, off"
                         :: "v"(lds_addr), "v"(gp) : "memory");
        }
        // --- build G slab in LDS (overlaps with in-flight async copies) ---
        for (int i = tid; i < COUTP * KSLAB; i += 256) {
            const int o = i / KSLAB;
            const int kc = i % KSLAB;
            const int kl = kc / CIN;
            const int c = kc % CIN;
            float s = 0.f;
            if (o < COUT) {
                const float* wr = &s_w[(k0 + kl) * 16];
                const float* lr = &lw[o * (CIN * 16) + c * 16];
                #pragma unroll
                for (int e = 0; e < 16; ++e) s += wr[e] * lr[e];
            }
            s_G[o * KSLAB + kc] = s;
        }
        asm volatile("s_wait_asynccnt 0" ::: "memory");
        __syncthreads();
        if (wact) {
            #pragma unroll
            for (int kk = 0; kk < KSLAB / 4; ++kk) {
                const int kb = kk * 4 + khalf;
                v2f a, bv;
                a.x  = s_G[rowo * KSLAB + kb];
                a.y  = s_G[rowo * KSLAB + kb + 1];
                bv.x = s_nv[kb * 65 + colb];
                bv.y = s_nv[(kb + 1) * 65 + colb];
                acc = __builtin_amdgcn_wmma_f32_16x16x4_f32(
                    false, a, false, bv, (short)0, acc, false, false);
            }
        }
        __syncthreads();
    }

    if (wact) {
        const int mb = mt * 16 + (lane >> 4) * 8;  // D: M = r + 8*(lane/16)
        #pragma unroll
        for (int r = 0; r < 8; ++r) {
            const int o = mb + r;
            if (o < COUT) {
                float v = (acc[r] + lb[o]) * (1.0f / 25.0f);
                if (ACT == 1) v = swishf(v);
                if (ACT == 2 && o == 1) v = softplusf(v);   // sigma channel
                fout[((size_t)colb * NPTS + n) * COUT + o] = v;
            }
        }
    }
}

// ------------------------- rbf readout: mu, diag(Sigma) --------------------
__global__ void k_musig(const float* __restrict__ tgt, const float* __restrict__ kp,
                        const float* __restrict__ ff, float* __restrict__ mu,
                        float* __restrict__ Sig) {
    int t = blockIdx.x * blockDim.x + threadIdx.x;
    if (t >= BATCH * NPTS) return;
    int b = t / NPTS, tt = t % NPTS;
    float tx = tgt[(size_t)t * 2 + 0], ty = tgt[(size_t)t * 2 + 1];
    float ls = kp[2], os = kp[3];
    float sc = -0.5f / (ls * ls);
    const float* fb = ff + (size_t)b * NPTS * 2;
    float m = 0.f, sg = 0.f;
    for (int n = 0; n < NPTS; ++n) {
        float dx = tx - gcoord(n / 28);
        float dy = ty - gcoord(n % 28);
        float k = os * __expf(sc * (dx * dx + dy * dy));
        m  += k * fb[n * 2 + 0];
        sg += k * fb[n * 2 + 1];            // softplus already applied in layer 4
    }
    mu[t] = m;
    Sig[(size_t)t * NPTS + tt] = sg;        // diagonal element [b, tt, tt]
}

// ---------------------------------------------------------------------------
extern "C" void kernel_launch(void* const* d_in, const int* in_sizes, int n_in,
                              void* d_out, int out_size, void* d_ws, size_t ws_size,
                              hipStream_t stream) {
    // ---- resolve input flattening order (detected from leaf sizes) ----
    // slots: 0=w1 1=b1 2=w2 3=b2 4=w3 5=b3 6=lw 7=lb within each layer group
    int ord_ins[8]    = {0, 1, 2, 3, 4, 5, 6, 7};              // insertion order
    int ord_sorted[8] = {5, 0, 6, 1, 7, 2, 4, 3};              // keys sorted: b1,b2,b3,lb,lw,w1,w2,w3
    const int* ord;
    int cnnBase, iCtxC, iCtxV, iTgt, iKp;
    if (in_sizes[0] == BATCH * NCTX * 2) {                     // top-level insertion order
        iCtxC = 0; iCtxV = 1; iTgt = 3; cnnBase = 4; iKp = 36;
        ord = (in_sizes[4] == 64) ? ord_ins : ord_sorted;      // w1 has 64 elems, b1 has 32
    } else {                                                   // fully sorted pytree
        cnnBase = 0; iCtxC = 32; iCtxV = 34; iKp = 35; iTgt = 36;
        ord = ord_sorted;
    }

    const float* LW[4]; const float* LB[4];
    MLP4 P;
    for (int l = 0; l < 4; ++l) {
        int base = cnnBase + l * 8;
        P.w1[l] = (const float*)d_in[base + ord[0]];
        P.b1[l] = (const float*)d_in[base + ord[1]];
        P.w2[l] = (const float*)d_in[base + ord[2]];
        P.b2[l] = (const float*)d_in[base + ord[3]];
        P.w3[l] = (const float*)d_in[base + ord[4]];
        P.b3[l] = (const float*)d_in[base + ord[5]];
        LW[l]   = (const float*)d_in[base + ord[6]];
        LB[l]   = (const float*)d_in[base + ord[7]];
    }
    const float* ctx_c = (const float*)d_in[iCtxC];
    const float* ctx_v = (const float*)d_in[iCtxV];
    const float* tgt   = (const float*)d_in[iTgt];
    const float* kp    = (const float*)d_in[iKp];

    // ---- workspace layout (floats); total ~18.1 MB ----
    int*   ws_idx  = (int*)d_ws;                               // 784*25
    float* ws_offs = (float*)d_ws + 19600;                     // 784*25*2
    float* ws_wmat = ws_offs + 39200;                          // 4*784*25*16
    float* fA      = ws_wmat + 4 * NPTS * NBHD_K * 16;         // 64*784*32
    float* fB      = fA + BATCH * NPTS * 32;                   // 64*784*32

    float* mu  = (float*)d_out;                                // (64,784)
    float* Sig = mu + BATCH * NPTS;                            // (64,784,784)

    // Sigma off-diagonals must be zero every call (harness poisons once).
    k_zero<<<4096, 256, 0, stream>>>((float4*)Sig,
                                     (long)BATCH * NPTS * NPTS / 4);
    k_neighbors<<<(NPTS + 255) / 256, 256, 0, stream>>>(ws_idx, ws_offs);
    k_wmlp<<<(4 * NPTS * NBHD_K + 255) / 256, 256, 0, stream>>>(ws_offs, ws_wmat, P);
    k_h<<<(BATCH * NPTS) / 256, 256, 0, stream>>>(ctx_c, ctx_v, kp, fA);

    const int WSTR = NPTS * NBHD_K * 16;
    k_pointconv< 4, 16, 1><<<NPTS, 256, 0, stream>>>(fA, fB, ws_idx, ws_wmat + 0 * WSTR, LW[0], LB[0]);
    k_pointconv<16, 32, 1><<<NPTS, 256, 0, stream>>>(fB, fA, ws_idx, ws_wmat + 1 * WSTR, LW[1], LB[1]);
    k_pointconv<32, 16, 1><<<NPTS, 256, 0, stream>>>(fA, fB, ws_idx, ws_wmat + 2 * WSTR, LW[2], LB[2]);
    k_pointconv<16,  2, 2><<<NPTS, 256, 0, stream>>>(fB, fA, ws_idx, ws_wmat + 3 * WSTR, LW[3], LB[3]);

    k_musig<<<(BATCH * NPTS) / 256, 256, 0, stream>>>(tgt, kp, fA, mu, Sig);
}